// AutoencoderHybrid_65481071408310
// MI455X (gfx1250) — compile-verified
//
#include <hip/hip_runtime.h>
#include <hip/hip_bf16.h>

// ---------------------------------------------------------------------------
// Quantum autoencoder hybrid: 8-qubit statevector sim (one wave32 per batch
// element, state in registers; cross-lane gates via DPP8/permlane16/swizzle)
// + WMMA f32 MLP head (V_WMMA_F32_16X16X4_F32).
// ---------------------------------------------------------------------------

#define NQ 8
#define REPS 4
#define BATCH 65536
#define ELEMS_PER_BLOCK 16   // 16 waves * 32 lanes = 512 threads

typedef __attribute__((ext_vector_type(2))) float v2f;
typedef __attribute__((ext_vector_type(8))) float v8f;

// ---------------- fast xor-shuffles ----------------------------------------
// masks 1/2/4: DPP8 (pure VALU, no LDS pipe, no dscnt wait)
// mask  8    : v_permlane16_b32 with constant xor-8 selector (row of 16)
// mask 16    : ds_swizzle via __shfl_xor (no wave32 half-swap VALU op)

constexpr unsigned dpp8_sel_xor(int m) {
  unsigned s = 0;
  for (int i = 0; i < 8; ++i) s |= (unsigned)((i ^ m) & 7) << (3 * i);
  return s;
}

template <int M>
__device__ __forceinline__ float shx(float v) {
  if constexpr (M == 1 || M == 2 || M == 4) {
    const int r = __builtin_amdgcn_mov_dpp8(__float_as_int(v), dpp8_sel_xor(M));
    return __int_as_float(r);
  } else if constexpr (M == 8) {
#if __has_builtin(__builtin_amdgcn_permlane16)
    const unsigned x = (unsigned)__float_as_int(v);
    const unsigned r =
        __builtin_amdgcn_permlane16(x, x, 0xFEDCBA98u, 0x76543210u, false, false);
    return __int_as_float((int)r);
#else
    return __shfl_xor(v, 8);
#endif
  } else {
    return __shfl_xor(v, M);
  }
}

// ---------------- gate helpers (all wirings compile-time) -------------------
// Amplitude index n (8 bits): bits 0..2 = per-lane register index (wires 0..2),
// bits 3..7 = lane id bits (wires 3..7).

template <int Q>
__device__ __forceinline__ void apply_rx(float (&sr)[8], float (&si)[8],
                                         float c, float s, int lane) {
  // U = [[c, -i s], [-i s, c]]  (symmetric): new = c*own - i*s*other
  if constexpr (Q < 3) {
    constexpr int m = 1 << Q;
#pragma unroll
    for (int a = 0; a < 8; ++a) {
      if (!(a & m)) {
        const int b = a | m;
        const float r0 = sr[a], i0 = si[a], r1 = sr[b], i1 = si[b];
        sr[a] = c * r0 + s * i1;
        si[a] = c * i0 - s * r1;
        sr[b] = c * r1 + s * i0;
        si[b] = c * i1 - s * r0;
      }
    }
  } else {
    constexpr int lm = 1 << (Q - 3);
#pragma unroll
    for (int a = 0; a < 8; ++a) {
      const float otr = shx<lm>(sr[a]);
      const float oti = shx<lm>(si[a]);
      const float mr = sr[a], mi = si[a];
      sr[a] = c * mr + s * oti;
      si[a] = c * mi - s * otr;
    }
  }
}

template <int Q>
__device__ __forceinline__ void apply_rot(float (&sr)[8], float (&si)[8],
                                          float u00r, float u00i, float u01r, float u01i,
                                          float u10r, float u10i, float u11r, float u11i,
                                          int lane) {
  if constexpr (Q < 3) {
    constexpr int m = 1 << Q;
#pragma unroll
    for (int a = 0; a < 8; ++a) {
      if (!(a & m)) {
        const int b = a | m;
        const float r0 = sr[a], i0 = si[a], r1 = sr[b], i1 = si[b];
        sr[a] = u00r * r0 - u00i * i0 + u01r * r1 - u01i * i1;
        si[a] = u00r * i0 + u00i * r0 + u01r * i1 + u01i * r1;
        sr[b] = u10r * r0 - u10i * i0 + u11r * r1 - u11i * i1;
        si[b] = u10r * i0 + u10i * r0 + u11r * i1 + u11i * r1;
      }
    }
  } else {
    constexpr int lm = 1 << (Q - 3);
    const bool hi = (lane & lm) != 0;
    // own coeff: u00 (bit=0) / u11 (bit=1); other coeff: u01 / u10
    const float uar = hi ? u11r : u00r, uai = hi ? u11i : u00i;
    const float ubr = hi ? u10r : u01r, ubi = hi ? u10i : u01i;
#pragma unroll
    for (int a = 0; a < 8; ++a) {
      const float otr = shx<lm>(sr[a]);
      const float oti = shx<lm>(si[a]);
      const float mr = sr[a], mi = si[a];
      sr[a] = uar * mr - uai * mi + ubr * otr - ubi * oti;
      si[a] = uar * mi + uai * mr + ubr * oti + ubi * otr;
    }
  }
}

template <int C, int T>
__device__ __forceinline__ void apply_cnot(float (&sr)[8], float (&si)[8], int lane) {
  if constexpr (C < 3 && T < 3) {
    constexpr int cm = 1 << C, tm = 1 << T;
#pragma unroll
    for (int a = 0; a < 8; ++a) {
      if ((a & cm) && !(a & tm)) {
        const int b = a | tm;
        float t0 = sr[a]; sr[a] = sr[b]; sr[b] = t0;
        float t1 = si[a]; si[a] = si[b]; si[b] = t1;
      }
    }
  } else if constexpr (C < 3 && T >= 3) {
    constexpr int cm = 1 << C, lm = 1 << (T - 3);
#pragma unroll
    for (int a = 0; a < 8; ++a) {
      if (a & cm) {
        sr[a] = shx<lm>(sr[a]);
        si[a] = shx<lm>(si[a]);
      }
    }
  } else if constexpr (C >= 3 && T < 3) {
    constexpr int lm = 1 << (C - 3), tm = 1 << T;
    const bool ctl = (lane & lm) != 0;
#pragma unroll
    for (int a = 0; a < 8; ++a) {
      if (!(a & tm)) {
        const int b = a | tm;
        const float ar = sr[a], ai = si[a], br = sr[b], bi = si[b];
        sr[a] = ctl ? br : ar;  si[a] = ctl ? bi : ai;
        sr[b] = ctl ? ar : br;  si[b] = ctl ? ai : bi;
      }
    }
  } else {
    constexpr int lm = 1 << (C - 3), tm = 1 << (T - 3);
    const bool ctl = (lane & lm) != 0;
#pragma unroll
    for (int a = 0; a < 8; ++a) {
      const float shr = shx<tm>(sr[a]);
      const float shi = shx<tm>(si[a]);
      sr[a] = ctl ? shr : sr[a];
      si[a] = ctl ? shi : si[a];
    }
  }
}

// ---------------- prep kernel: shared Rot(phi,theta,omega) matrices ---------
__global__ void qae_prep_rot(const float* __restrict__ qw, float* __restrict__ rm) {
  const int t = threadIdx.x;
  if (t < REPS * NQ) {
    const float phi = qw[t * 3 + 0];
    const float th  = qw[t * 3 + 1];
    const float om  = qw[t * 3 + 2];
    const float c = cosf(0.5f * th), s = sinf(0.5f * th);
    const float a = 0.5f * (phi + om), b = 0.5f * (phi - om);
    const float ca = cosf(a), sa = sinf(a), cb = cosf(b), sb = sinf(b);
    float* o = rm + t * 8;
    o[0] =  ca * c;  o[1] = -sa * c;   // m00 = exp(-i(phi+om)/2) c
    o[2] = -cb * s;  o[3] = -sb * s;   // m01 = -exp(+i(phi-om)/2) s
    o[4] =  cb * s;  o[5] = -sb * s;   // m10 =  exp(-i(phi-om)/2) s
    o[6] =  ca * c;  o[7] =  sa * c;   // m11 =  exp(+i(phi+om)/2) c
  }
}

// ---------------- main kernel ----------------------------------------------
#define ROT(L, I)                                                              \
  do {                                                                         \
    const float* m_ = rm + ((L) * 8 + (I)) * 8;                                \
    apply_rot<(I)>(sr, si, m_[0], m_[1], m_[2], m_[3], m_[4], m_[5], m_[6],    \
                   m_[7], lane);                                               \
  } while (0)
#define RLAYER(L) \
  ROT(L,0); ROT(L,1); ROT(L,2); ROT(L,3); ROT(L,4); ROT(L,5); ROT(L,6); ROT(L,7)
#define CLAYER(R)                                                 \
  apply_cnot<0, (0 + (R)) & 7>(sr, si, lane);                     \
  apply_cnot<1, (1 + (R)) & 7>(sr, si, lane);                     \
  apply_cnot<2, (2 + (R)) & 7>(sr, si, lane);                     \
  apply_cnot<3, (3 + (R)) & 7>(sr, si, lane);                     \
  apply_cnot<4, (4 + (R)) & 7>(sr, si, lane);                     \
  apply_cnot<5, (5 + (R)) & 7>(sr, si, lane);                     \
  apply_cnot<6, (6 + (R)) & 7>(sr, si, lane);                     \
  apply_cnot<7, (7 + (R)) & 7>(sr, si, lane)

__launch_bounds__(512)
__global__ void qae_kernel(const float* __restrict__ x, const float* __restrict__ rm,
                           const float* __restrict__ w1, const float* __restrict__ b1g,
                           const float* __restrict__ w2, const float* __restrict__ b2g,
                           float* __restrict__ out) {
  __shared__ float zs[16][8];       // z rows for the 16 elements of this block
  __shared__ float B1[2][4][16];    // W1^T in WMMA B layout, 2 K-chunks
  __shared__ float B2m[4][16];      // W2^T in WMMA B layout
  __shared__ float Hs[16][4];       // hidden activations

  const int t = threadIdx.x;
  const int lane = t & 31;
  const int wid = t >> 5;

  // Stage the tiny weight panels (B-matrix layout: row k striped over lanes).
  if (t < 128) {
    const int c = t >> 6, k = (t >> 4) & 3, n = t & 15;
    B1[c][k][n] = (n < 4) ? w1[n * 8 + c * 4 + k] : 0.0f;   // W1^T[c*4+k][n]
  } else if (t < 192) {
    const int k = (t >> 4) & 3, n = t & 15;
    B2m[k][n] = (n < 8) ? w2[n * 4 + k] : 0.0f;             // W2^T[k][n]
  }

  const int e = blockIdx.x * ELEMS_PER_BLOCK + wid;
  const int es = __builtin_amdgcn_readfirstlane(e);         // force SGPR index
  const float* xe = x + (size_t)es * 8;

  // |0...0>
  float sr[8], si[8];
#pragma unroll
  for (int a = 0; a < 8; ++a) { sr[a] = 0.0f; si[a] = 0.0f; }
  sr[0] = (lane == 0) ? 1.0f : 0.0f;

  // ---- encode: RX(x_i) on wire i ----
  {
    float th, c, s;
    th = xe[0]; c = __cosf(0.5f * th); s = __sinf(0.5f * th); apply_rx<0>(sr, si, c, s, lane);
    th = xe[1]; c = __cosf(0.5f * th); s = __sinf(0.5f * th); apply_rx<1>(sr, si, c, s, lane);
    th = xe[2]; c = __cosf(0.5f * th); s = __sinf(0.5f * th); apply_rx<2>(sr, si, c, s, lane);
    th = xe[3]; c = __cosf(0.5f * th); s = __sinf(0.5f * th); apply_rx<3>(sr, si, c, s, lane);
    th = xe[4]; c = __cosf(0.5f * th); s = __sinf(0.5f * th); apply_rx<4>(sr, si, c, s, lane);
    th = xe[5]; c = __cosf(0.5f * th); s = __sinf(0.5f * th); apply_rx<5>(sr, si, c, s, lane);
    th = xe[6]; c = __cosf(0.5f * th); s = __sinf(0.5f * th); apply_rx<6>(sr, si, c, s, lane);
    th = xe[7]; c = __cosf(0.5f * th); s = __sinf(0.5f * th); apply_rx<7>(sr, si, c, s, lane);
  }

  // ---- entangling layers: r = l % 7 + 1 ----
  RLAYER(0); CLAYER(1);
  RLAYER(1); CLAYER(2);
  RLAYER(2); CLAYER(3);
  RLAYER(3); CLAYER(4);

  // ---- Pauli-Z expectations ----
  float pr[8];
  float tot = 0.0f;
#pragma unroll
  for (int a = 0; a < 8; ++a) {
    pr[a] = sr[a] * sr[a] + si[a] * si[a];
    tot += pr[a];
  }
  float zv[8];
  zv[0] = pr[0] - pr[1] + pr[2] - pr[3] + pr[4] - pr[5] + pr[6] - pr[7];
  zv[1] = pr[0] + pr[1] - pr[2] - pr[3] + pr[4] + pr[5] - pr[6] - pr[7];
  zv[2] = pr[0] + pr[1] + pr[2] + pr[3] - pr[4] - pr[5] - pr[6] - pr[7];
  zv[3] = (lane & 1)  ? -tot : tot;
  zv[4] = (lane & 2)  ? -tot : tot;
  zv[5] = (lane & 4)  ? -tot : tot;
  zv[6] = (lane & 8)  ? -tot : tot;
  zv[7] = (lane & 16) ? -tot : tot;
#pragma unroll
  for (int k = 0; k < 8; ++k) zv[k] += shx<16>(zv[k]);
#pragma unroll
  for (int k = 0; k < 8; ++k) zv[k] += shx<8>(zv[k]);
#pragma unroll
  for (int k = 0; k < 8; ++k) zv[k] += shx<4>(zv[k]);
#pragma unroll
  for (int k = 0; k < 8; ++k) zv[k] += shx<2>(zv[k]);
#pragma unroll
  for (int k = 0; k < 8; ++k) zv[k] += shx<1>(zv[k]);

  if (lane == 0) {
#pragma unroll
    for (int k = 0; k < 8; ++k) zs[wid][k] = zv[k];
  }
  __syncthreads();

  // ---- MLP head: wave 0 does 16 elements via V_WMMA_F32_16X16X4_F32 ----
  if (wid == 0) {
    const int col = lane & 15;   // WMMA N / A-matrix M
    const int half = lane >> 4;  // selects K pair {0,1} vs {2,3}

    // GEMM1: D = Z(16x8) @ W1^T(8x4) over two K=4 chunks
    v8f acc = {0.f, 0.f, 0.f, 0.f, 0.f, 0.f, 0.f, 0.f};
#pragma unroll
    for (int c2 = 0; c2 < 2; ++c2) {
      v2f a, b;
      a.x = zs[col][c2 * 4 + half * 2 + 0];
      a.y = zs[col][c2 * 4 + half * 2 + 1];
      b.x = B1[c2][half * 2 + 0][col];
      b.y = B1[c2][half * 2 + 1][col];
      acc = __builtin_amdgcn_wmma_f32_16x16x4_f32(false, a, false, b,
                                                  (short)0, acc, false, false);
    }
    // bias + ReLU, stage H (valid cols 0..3) to LDS in A-matrix feed order
    const float b1v = (col < 4) ? b1g[col] : 0.0f;
#pragma unroll
    for (int r = 0; r < 8; ++r) {
      float h = acc[r] + b1v;
      h = h > 0.0f ? h : 0.0f;
      if (col < 4) Hs[r + 8 * half][col] = h;
    }
    __asm__ volatile("s_wait_dscnt 0" ::: "memory");  // intra-wave LDS RAW

    // GEMM2: OUT = H(16x4) @ W2^T(4x8), single K=4 chunk
    v2f a2, b2x;
    a2.x = Hs[col][half * 2 + 0];
    a2.y = Hs[col][half * 2 + 1];
    b2x.x = B2m[half * 2 + 0][col];
    b2x.y = B2m[half * 2 + 1][col];
    v8f acc2 = {0.f, 0.f, 0.f, 0.f, 0.f, 0.f, 0.f, 0.f};
    acc2 = __builtin_amdgcn_wmma_f32_16x16x4_f32(false, a2, false, b2x,
                                                 (short)0, acc2, false, false);

    const float b2v = (col < 8) ? b2g[col] : 0.0f;
    if (col < 8) {
      const size_t base = (size_t)blockIdx.x * ELEMS_PER_BLOCK * 8;
#pragma unroll
      for (int r = 0; r < 8; ++r) {
        out[base + (size_t)(r + 8 * half) * 8 + col] = acc2[r] + b2v;
      }
    }
  }
}

// ---------------- launch ----------------------------------------------------
extern "C" void kernel_launch(void* const* d_in, const int* in_sizes, int n_in,
                              void* d_out, int out_size, void* d_ws, size_t ws_size,
                              hipStream_t stream) {
  const float* x  = (const float*)d_in[0];
  const float* qw = (const float*)d_in[1];
  const float* w1 = (const float*)d_in[2];
  const float* b1 = (const float*)d_in[3];
  const float* w2 = (const float*)d_in[4];
  const float* b2 = (const float*)d_in[5];
  float* out = (float*)d_out;
  float* rm = (float*)d_ws;  // 32 gates * 8 floats = 1 KB scratch

  qae_prep_rot<<<1, 32, 0, stream>>>(qw, rm);
  qae_kernel<<<BATCH / ELEMS_PER_BLOCK, ELEMS_PER_BLOCK * 32, 0, stream>>>(
      x, rm, w1, b1, w2, b2, out);
}